// KernelAttention_28149215658653
// MI455X (gfx1250) — compile-verified
//
#include <hip/hip_runtime.h>

// ---------------------------------------------------------------------------
// Linear attention (elu+1 feature map) for MI455X / gfx1250, fp32 end-to-end.
//   pass1: kv[d,v] = sum_n phi_k[n,d] * V[n,v]  (WMMA f32 16x16x4, LDS transpose)
//   pass2: out[n,v] = (phi_q[n,:] @ kv) / (phi_q[n,:] . k_one + eps)
// Q/K/V and kv loads vectorized to b128 (bandwidth-bound: ~256MB @ 23.3TB/s).
// ---------------------------------------------------------------------------

typedef __attribute__((ext_vector_type(2))) float v2f;
typedef __attribute__((ext_vector_type(8))) float v8f;

#define B_ 4
#define H_ 16
#define S_ 4096
#define D_ 64
#define BH_ (B_ * H_)
#define EPS_ 1e-6f

#define KV_STRIDE 4160        // 64*64 kv + 64 k_one floats per head
#define SCHUNK 512            // rows of S handled per pass-1 block
#define P2_ROWS 128           // rows of S handled per pass-2 block

#define QSTRIDE 68            // lds_q row stride (mult of 4 -> b128; odd/4 -> conflict-free column gather)
#define KVSTRIDE_L 72         // lds_kv row stride (halves of wave hit disjoint bank sets)

__device__ __forceinline__ float phi_f(float x) {
    // elu(x)+1 == (x > 0) ? x + 1 : exp(x)
    return x > 0.0f ? x + 1.0f : __expf(x);
}

__global__ void la_zero_ws(float4* __restrict__ ws, int n4) {
    int i = blockIdx.x * blockDim.x + threadIdx.x;
    if (i < n4) ws[i] = make_float4(0.f, 0.f, 0.f, 0.f);
}

// ---------------------------------------------------------------------------
// Pass 1: per (head, S-chunk) block accumulates kv = phi_k^T @ V and k_one.
// grid = (BH, S/SCHUNK), block = 256 (8 waves). Each wave owns 2 of the 16
// 16x16 output tiles of the 64x64 kv matrix.
// ---------------------------------------------------------------------------
__global__ __launch_bounds__(256) void la_pass1_kvterm(
    const float* __restrict__ K, const float* __restrict__ V,
    const float* __restrict__ mask, float* __restrict__ ws) {

    const int bh = blockIdx.x;
    const int chunk = blockIdx.y;
    const int b = bh / H_;
    const size_t base = (size_t)bh * S_ * D_ + (size_t)chunk * SCHUNK * D_;
    const float* __restrict__ Kp = K + base;
    const float* __restrict__ Vp = V + base;
    const float* __restrict__ Mp = mask + (size_t)b * S_ + (size_t)chunk * SCHUNK;

    __shared__ __align__(16) float lds_kT[D_][17]; // transposed phi_k slab [d][n_local]
    __shared__ __align__(16) float lds_v[16][64];  // V slab row-major [n_local][v]
    __shared__ float lds_kone[D_];

    const int tid  = threadIdx.x;
    const int lane = tid & 31;
    const int wid  = tid >> 5;
    const int l15  = lane & 15;
    const int half = lane >> 4;        // wave32: which half of the wave

    const int dTile0 = wid >> 2;       // 0..1
    const int dTile1 = dTile0 + 2;     // 2..3
    const int vTile  = wid & 3;        // 0..3

    // b128 loader mapping: one row-segment of 4 floats per thread per slab
    const int nl  = tid >> 4;          // 0..15 (row within slab)
    const int col = (tid & 15) * 4;    // 0,4,...,60

    v8f c0 = {}; v8f c1 = {};
    float kone_acc[4] = {0.f, 0.f, 0.f, 0.f};

    if (tid < D_) lds_kone[tid] = 0.0f;

    for (int n0 = 0; n0 < SCHUNK; n0 += 16) {
        __syncthreads();               // previous slab fully consumed
        const float4 kd = *(const float4*)&Kp[(size_t)(n0 + nl) * D_ + col];
        const float4 vv = *(const float4*)&Vp[(size_t)(n0 + nl) * D_ + col];
        const float m = Mp[n0 + nl];
        const float pk0 = phi_f(kd.x) * m;
        const float pk1 = phi_f(kd.y) * m;
        const float pk2 = phi_f(kd.z) * m;
        const float pk3 = phi_f(kd.w) * m;
        lds_kT[col + 0][nl] = pk0;     // transpose scatter (conflict-free @ stride 17)
        lds_kT[col + 1][nl] = pk1;
        lds_kT[col + 2][nl] = pk2;
        lds_kT[col + 3][nl] = pk3;
        kone_acc[0] += pk0; kone_acc[1] += pk1;
        kone_acc[2] += pk2; kone_acc[3] += pk3;
        *(float4*)&lds_v[nl][col] = vv;   // one ds_store_b128
        __syncthreads();

        #pragma unroll
        for (int kk = 0; kk < 16; kk += 4) {
            const int kc = kk + half * 2;            // A/B K-index per lane half
            v2f a0, a1, bb;
            a0.x = lds_kT[dTile0 * 16 + l15][kc];
            a0.y = lds_kT[dTile0 * 16 + l15][kc + 1];
            a1.x = lds_kT[dTile1 * 16 + l15][kc];
            a1.y = lds_kT[dTile1 * 16 + l15][kc + 1];
            bb.x = lds_v[kc][vTile * 16 + l15];
            bb.y = lds_v[kc + 1][vTile * 16 + l15];
            c0 = __builtin_amdgcn_wmma_f32_16x16x4_f32(false, a0, false, bb,
                                                       (short)0, c0, false, false);
            c1 = __builtin_amdgcn_wmma_f32_16x16x4_f32(false, a1, false, bb,
                                                       (short)0, c1, false, false);
        }
    }

    // k_one: 16 partial sums per d column -> LDS atomics -> global atomics
    #pragma unroll
    for (int j = 0; j < 4; ++j) atomicAdd(&lds_kone[col + j], kone_acc[j]);
    __syncthreads();

    float* __restrict__ wkv = ws + (size_t)bh * KV_STRIDE;
    if (tid < D_) atomicAdd(&wkv[4096 + tid], lds_kone[tid]);

    // kv tiles: D-matrix VGPR j holds rows M=j (lanes 0-15) / M=j+8 (lanes 16-31)
    #pragma unroll
    for (int j = 0; j < 8; ++j) {
        const int r0 = dTile0 * 16 + j + 8 * half;
        const int r1 = dTile1 * 16 + j + 8 * half;
        const int cidx = vTile * 16 + l15;
        atomicAdd(&wkv[r0 * 64 + cidx], c0[j]);
        atomicAdd(&wkv[r1 * 64 + cidx], c1[j]);
    }
}

// ---------------------------------------------------------------------------
// Pass 2: out = phi_q @ kv, normalized by phi_q . k_one + eps.
// grid = (BH, S/P2_ROWS), block = 256; each wave computes 16 output rows.
// ---------------------------------------------------------------------------
__global__ __launch_bounds__(256) void la_pass2_output(
    const float* __restrict__ Q, const float* __restrict__ ws,
    float* __restrict__ out) {

    const int bh = blockIdx.x;
    const int rblk = blockIdx.y;
    const size_t qbase = (size_t)bh * S_ * D_ + (size_t)rblk * P2_ROWS * D_;
    const float* __restrict__ Qp = Q + qbase;
    float* __restrict__ Op = out + qbase;
    const float* __restrict__ wkv = ws + (size_t)bh * KV_STRIDE;

    __shared__ __align__(16) float lds_kv[D_][KVSTRIDE_L];
    __shared__ float lds_kone[D_];
    __shared__ __align__(16) float lds_q[P2_ROWS][QSTRIDE];

    const int tid  = threadIdx.x;
    const int lane = tid & 31;
    const int wid  = tid >> 5;
    const int l15  = lane & 15;
    const int half = lane >> 4;

    // b128 loaders: kv = 1024 float4 (4/thread), q = 2048 float4 (8/thread)
    #pragma unroll
    for (int i = 0; i < 4; ++i) {
        const int idx = tid + 256 * i;
        const int r = idx >> 4;
        const int c = (idx & 15) * 4;
        *(float4*)&lds_kv[r][c] = *(const float4*)&wkv[r * 64 + c];
    }
    if (tid < D_) lds_kone[tid] = wkv[4096 + tid];

    #pragma unroll
    for (int i = 0; i < 8; ++i) {
        const int idx = tid + 256 * i;
        const int r = idx >> 4;
        const int c = (idx & 15) * 4;
        float4 qv = *(const float4*)&Qp[(size_t)r * D_ + c];
        qv.x = phi_f(qv.x * 0.125f);    // 1/sqrt(64)
        qv.y = phi_f(qv.y * 0.125f);
        qv.z = phi_f(qv.z * 0.125f);
        qv.w = phi_f(qv.w * 0.125f);
        *(float4*)&lds_q[r][c] = qv;
    }
    __syncthreads();

    const int mb = wid * 16;                    // this wave's row base
    v8f acc[4] = {{}, {}, {}, {}};

    #pragma unroll
    for (int kk = 0; kk < 64; kk += 4) {
        const int kc = kk + half * 2;
        v2f a;
        a.x = lds_q[mb + l15][kc];
        a.y = lds_q[mb + l15][kc + 1];
        #pragma unroll
        for (int vt = 0; vt < 4; ++vt) {
            v2f bfrag;
            bfrag.x = lds_kv[kc][vt * 16 + l15];
            bfrag.y = lds_kv[kc + 1][vt * 16 + l15];
            acc[vt] = __builtin_amdgcn_wmma_f32_16x16x4_f32(false, a, false, bfrag,
                                                            (short)0, acc[vt], false, false);
        }
    }

    // normalizer: lane half sums 32 of the 64 d's for row mb+l15, xor-combine
    float part = 0.0f;
    const int nrow = mb + l15;
    #pragma unroll
    for (int d = 0; d < 32; ++d) {
        const int dd = half * 32 + d;
        part += lds_q[nrow][dd] * lds_kone[dd];
    }
    part += __shfl_xor(part, 16, 32);
    const float inv = 1.0f / (part + EPS_);

    #pragma unroll
    for (int j = 0; j < 8; ++j) {
        const float rn = __shfl(inv, j + 8 * half, 32);  // norm of this VGPR's row
        const int r = mb + j + 8 * half;
        #pragma unroll
        for (int vt = 0; vt < 4; ++vt) {
            Op[(size_t)r * D_ + vt * 16 + l15] = acc[vt][j] * rn;
        }
    }
}

extern "C" void kernel_launch(void* const* d_in, const int* in_sizes, int n_in,
                              void* d_out, int out_size, void* d_ws, size_t ws_size,
                              hipStream_t stream) {
    const float* Q = (const float*)d_in[0];
    const float* K = (const float*)d_in[1];
    const float* V = (const float*)d_in[2];
    const float* M = (const float*)d_in[3];
    float* out = (float*)d_out;
    float* ws  = (float*)d_ws;

    const int nz4 = (BH_ * KV_STRIDE) / 4;   // KV_STRIDE is a multiple of 4
    la_zero_ws<<<(nz4 + 255) / 256, 256, 0, stream>>>((float4*)ws, nz4);
    la_pass1_kvterm<<<dim3(BH_, S_ / SCHUNK), 256, 0, stream>>>(K, V, M, ws);
    la_pass2_output<<<dim3(BH_, S_ / P2_ROWS), 256, 0, stream>>>(Q, ws, out);
}